// FastFoodProjection_29549374997209
// MI455X (gfx1250) — compile-verified
//
#include <hip/hip_runtime.h>
#include <math.h>

// ---------------- problem constants ----------------
#define SIZE_LOG2 24
#define SIZE_N    (1 << SIZE_LOG2)      // 16,777,216
#define FLAT_N    12000000
#define D_INT     4096

typedef __attribute__((ext_vector_type(2))) float v2f;
typedef __attribute__((ext_vector_type(8))) float v8f;

// Skewed LDS layout: pitch 17 per 16 -> all three WMMA rounds are
// bank-conflict free (verified: round0 loads stride 17, rounds 1/2 stride 1).
#define LDS_PHYS 4352
__device__ __forceinline__ int skew(int j) { return j + (j >> 4); }

__device__ __forceinline__ float hsign(int m, int k) {
  return (__builtin_popcount(m & k) & 1) ? -1.0f : 1.0f;
}

// index of (axis value x, "other" index t) for the three radix-16 rounds of
// FWHT_4096 over j = a*256 + b*16 + c
__device__ __forceinline__ int fwht_idx(int round, int x, int t) {
  if (round == 0) return t * 16 + x;                          // axis c, t=(a,b)
  if (round == 1) return (t >> 4) * 256 + x * 16 + (t & 15);  // axis b, t=(a,c)
  return x * 256 + t;                                         // axis a, t=(b,c)
}

// 4096-point unnormalized FWHT over skewed LDS array, computed as three
// rounds of H16 x M(16x16) via V_WMMA_F32_16X16X4_F32 (exact f32 math).
// Requires blockDim.x == 256 (8 waves), all lanes active (EXEC all ones).
__device__ __forceinline__ void fwht4096_wmma(float* s) {
  const int tid  = threadIdx.x;
  const int lane = tid & 31;
  const int wave = tid >> 5;
  const int n    = lane & 15;   // also A-matrix row m and D-matrix column
  const int hk   = lane >> 4;   // K half / M half selector

  // H16 A-matrix chunks (16x4 each). ISA layout: A[m][k] at
  // lane = 16*(k>=2)+m, vgpr = k&1  ->  chunk kc: .x -> k=4kc+2hk, .y -> +1
  v2f A[4];
#pragma unroll
  for (int kc = 0; kc < 4; ++kc) {
    const int kb = kc * 4 + 2 * hk;
    A[kc].x = hsign(n, kb);
    A[kc].y = hsign(n, kb + 1);
  }

#pragma unroll
  for (int round = 0; round < 3; ++round) {
    __syncthreads();  // tiles partition t-space, so only inter-round syncs needed
#pragma unroll
    for (int ti = 0; ti < 2; ++ti) {
      const int tt = wave + ti * 8;     // tile id 0..15
      const int t  = tt * 16 + n;       // this lane's column
      v8f acc = {};
#pragma unroll
      for (int kc = 0; kc < 4; ++kc) {
        const int kb = kc * 4 + 2 * hk;
        v2f b;  // B 4x16: B[k][n] at lane = 16*(k>=2)+n, vgpr = k&1
        b.x = s[skew(fwht_idx(round, kb,     t))];
        b.y = s[skew(fwht_idx(round, kb + 1, t))];
        acc = __builtin_amdgcn_wmma_f32_16x16x4_f32(
            false, A[kc], false, b, (short)0, acc, false, false);
      }
      // D 16x16 f32: vgpr r holds row r (lanes 0-15) / row r+8 (lanes 16-31)
#pragma unroll
      for (int r = 0; r < 8; ++r) {
        const int m = r + 8 * hk;
        s[skew(fwht_idx(round, m, t))] = acc[r];
      }
    }
  }
  __syncthreads();
}

// 64-point FWHT fully in registers (6 unrolled stages).
__device__ __forceinline__ void fwht64(float v[64]) {
#pragma unroll
  for (int st = 1; st < 64; st <<= 1) {
#pragma unroll
    for (int i = 0; i < 64; ++i) {
      if ((i & st) == 0) {
        const float a = v[i], b = v[i + st];
        v[i]      = a + b;
        v[i + st] = a - b;
      }
    }
  }
}

// ---------------- Kernel A: y = FWHT_4096(theta * B[:4096]) ----------------
__global__ __launch_bounds__(256) void k_seed(const float* __restrict__ theta,
                                              const float* __restrict__ Bv,
                                              float* __restrict__ y) {
  __shared__ float s[LDS_PHYS];
  const int tid = threadIdx.x;
#pragma unroll
  for (int r = 0; r < 16; ++r) {
    const int j = tid + 256 * r;
    s[skew(j)] = theta[j] * Bv[j];
  }
  fwht4096_wmma(s);
#pragma unroll
  for (int r = 0; r < 16; ++r) {
    const int j = tid + 256 * r;
    y[j] = s[skew(j)];
  }
}

// ------- Kernel B: z = FWHT_low12( y[Pi & 4095] * G ), + partial sum(G^2) ----
__global__ __launch_bounds__(256) void k_gather_fwht(const int*   __restrict__ Pi,
                                                     const float* __restrict__ G,
                                                     const float* __restrict__ y,
                                                     float* __restrict__ z,
                                                     float* __restrict__ partials) {
  __shared__ float ys[D_INT];
  __shared__ float s[LDS_PHYS];
  __shared__ float red[256];
  const int tid = threadIdx.x;
  const int blk = blockIdx.x;

#pragma unroll
  for (int r = 0; r < 16; ++r) ys[tid + 256 * r] = y[tid + 256 * r];  // L2-hot
  __syncthreads();

  float acc = 0.0f;
#pragma unroll
  for (int r = 0; r < 16; ++r) {
    const int i = tid + 256 * r;
    const int j = (blk << 12) + i;
    const float g = G[j];
    acc += g * g;
    s[skew(i)] = ys[Pi[j] & (D_INT - 1)] * g;
  }
  fwht4096_wmma(s);
#pragma unroll
  for (int r = 0; r < 16; ++r) {
    const int i = tid + 256 * r;
    z[(blk << 12) + i] = s[skew(i)];
  }

  // deterministic block reduction of sum(G^2)
  red[tid] = acc;
  __syncthreads();
#pragma unroll
  for (int off = 128; off > 0; off >>= 1) {
    if (tid < off) red[tid] += red[tid + off];
    __syncthreads();
  }
  if (tid == 0) partials[blk] = red[0];
}

// ---------- Kernel C: deterministic reduce 4096 partials -> sumG2 ----------
__global__ __launch_bounds__(256) void k_reduce(const float* __restrict__ partials,
                                                float* __restrict__ sumG2) {
  __shared__ float red[256];
  const int tid = threadIdx.x;
  float a = 0.0f;
#pragma unroll
  for (int r = 0; r < 16; ++r) a += partials[tid * 16 + r];
  red[tid] = a;
  __syncthreads();
#pragma unroll
  for (int off = 128; off > 0; off >>= 1) {
    if (tid < off) red[tid] += red[tid + off];
    __syncthreads();
  }
  if (tid == 0) sumG2[0] = red[0];
}

// --------- Kernel P1: 64-pt FWHT at stride 4096 (stages 12..17) ------------
__global__ __launch_bounds__(256) void k_mid(float* __restrict__ z) {
  const int T    = blockIdx.x * 256 + threadIdx.x;        // 2^18 threads
  const int base = (T >> 12) * 262144 + (T & 4095);       // (hi, c)
  float v[64];
#pragma unroll
  for (int k = 0; k < 64; ++k) v[k] = z[base + k * 4096];
  fwht64(v);
#pragma unroll
  for (int k = 0; k < 64; ++k) z[base + k * 4096] = v[k];
}

// --- Kernel P2: 64-pt FWHT at stride 2^18 (stages 18..23) + scale + trunc ---
__global__ __launch_bounds__(256) void k_top(const float* __restrict__ z,
                                             const float* __restrict__ sumG2,
                                             float* __restrict__ out) {
  const int T = blockIdx.x * 256 + threadIdx.x;           // 2^18 threads
  float v[64];
#pragma unroll
  for (int k = 0; k < 64; ++k) v[k] = z[T + k * 262144];
  fwht64(v);
  const float scale = 1.0f / sqrtf((float)FLAT_N * sumG2[0]);
#pragma unroll
  for (int k = 0; k < 64; ++k) {
    const int j = T + k * 262144;
    if (j < FLAT_N) out[j] = v[k] * scale;
  }
}

// ---------------------------------------------------------------------------
extern "C" void kernel_launch(void* const* d_in, const int* in_sizes, int n_in,
                              void* d_out, int out_size, void* d_ws, size_t ws_size,
                              hipStream_t stream) {
  const float* theta = (const float*)d_in[0];   // 4096 f32
  const float* Bv    = (const float*)d_in[1];   // SIZE f32
  const float* G     = (const float*)d_in[2];   // SIZE f32
  const int*   Pi    = (const int*)  d_in[3];   // SIZE i32
  float* out = (float*)d_out;                   // FLAT f32

  float* ws       = (float*)d_ws;
  float* z        = ws;                         // SIZE floats (64 MiB)
  float* partials = ws + SIZE_N;                // 4096 floats
  float* sumG2    = ws + SIZE_N + 4096;         // 1 float
  float* y        = ws + SIZE_N + 4096 + 64;    // 4096 floats

  k_seed<<<1, 256, 0, stream>>>(theta, Bv, y);
  k_gather_fwht<<<SIZE_N / 4096, 256, 0, stream>>>(Pi, G, y, z, partials);
  k_reduce<<<1, 256, 0, stream>>>(partials, sumG2);
  k_mid<<<(SIZE_N / 64) / 256, 256, 0, stream>>>(z);
  k_top<<<(SIZE_N / 64) / 256, 256, 0, stream>>>(z, sumG2, out);
}